// Net_29935922053254
// MI455X (gfx1250) — compile-verified
//
#include <hip/hip_runtime.h>
#include <hip/hip_bf16.h>

typedef __attribute__((ext_vector_type(2))) float v2f;
typedef __attribute__((ext_vector_type(8))) float v8f;

// ---------------------------------------------------------------------------
// zero-fill (agg buffers must be zeroed before scatter-add every call)
// ---------------------------------------------------------------------------
__global__ void zero_f32(float* __restrict__ p, long long n) {
  long long i = (long long)blockIdx.x * blockDim.x + threadIdx.x;
  long long stride = (long long)gridDim.x * blockDim.x;
  for (; i < n; i += stride) p[i] = 0.0f;
}

// ---------------------------------------------------------------------------
// h = x @ Wx.T + bx     (x: [N,IN], Wx: [9,IN], h: [N,9])
// ---------------------------------------------------------------------------
template <int IN>
__global__ void node_linear(const float* __restrict__ x,
                            const float* __restrict__ W,
                            const float* __restrict__ b,
                            float* __restrict__ h, int N) {
  int i = blockIdx.x * blockDim.x + threadIdx.x;
  if (i >= N) return;
  float xin[IN];
#pragma unroll
  for (int k = 0; k < IN; ++k) xin[k] = x[(long long)i * IN + k];
#pragma unroll
  for (int j = 0; j < 9; ++j) {
    float acc = b[j];
#pragma unroll
    for (int k = 0; k < IN; ++k) acc = fmaf(xin[k], W[j * IN + k], acc);
    h[(long long)i * 9 + j] = acc;
  }
}

// ---------------------------------------------------------------------------
// msg = h[src] * (edge_attr @ We.T + be); agg[dst] += msg   (scatter via
// native f32 global atomics; h/agg arrays are L2-resident at 54 MB each)
// ---------------------------------------------------------------------------
__global__ void edge_scatter(const int* __restrict__ ei,   // [2,E]
                             const float* __restrict__ ea, // [E,2]
                             const float* __restrict__ We, // [9,2]
                             const float* __restrict__ be, // [9]
                             const float* __restrict__ h,  // [N,9]
                             float* __restrict__ agg,      // [N,9]
                             int E) {
  int e = blockIdx.x * blockDim.x + threadIdx.x;
  if (e >= E) return;
  int src = ei[e];
  int dst = ei[(long long)E + e];
  float a0 = ea[2 * (long long)e];
  float a1 = ea[2 * (long long)e + 1];
  const float* hs = h + (long long)src * 9;
  float* ag = agg + (long long)dst * 9;
#pragma unroll
  for (int j = 0; j < 9; ++j) {
    float w = fmaf(a1, We[2 * j + 1], fmaf(a0, We[2 * j], be[j]));
    unsafeAtomicAdd(ag + j, hs[j] * w);
  }
}

// ---------------------------------------------------------------------------
// Shared helper: one wave computes D = agg[16x9] @ Wo.T[9x16pad] with
// V_WMMA_F32_16X16X4_F32 (3 K-chunks, K padded 9->12, branchless clamped
// loads + cndmask zeroing so no EXEC juggling).
// A-frag (16x4 f32): lane l -> row m=l&15, k-pair base kb=2*(l>>4).
// B-frag (4x16 f32): lane l -> col n=l&15, same kb.
// D (16x16 f32, 8 VGPRs): lane l, vgpr v -> row v+8*(l>>4), col l&15.
// ---------------------------------------------------------------------------
template <int OUTD>
__device__ inline v8f wmma_9in_tile(const float* __restrict__ agg,
                                    const float* __restrict__ Wo,
                                    long long base, int lane) {
  const int n  = lane & 15;
  const int kb = (lane >> 4) * 2;
  const int nn = (n < OUTD) ? n : (OUTD - 1);  // clamped, always in-bounds
  const long long row = base + (lane & 15);    // N%16==0 -> in range

  v8f acc = {};
#pragma unroll
  for (int c = 0; c < 3; ++c) {
    const int k0 = c * 4 + kb;
    const int k1 = k0 + 1;
    const int ka = (k0 < 9) ? k0 : 8;          // clamped indices
    const int kb2 = (k1 < 9) ? k1 : 8;

    float ax = agg[row * 9 + ka];
    float ay = agg[row * 9 + kb2];
    ax = (k0 < 9) ? ax : 0.0f;
    ay = (k1 < 9) ? ay : 0.0f;

    float bx = Wo[nn * 9 + ka];
    float by = Wo[nn * 9 + kb2];
    bx = (n < OUTD && k0 < 9) ? bx : 0.0f;
    by = (n < OUTD && k1 < 9) ? by : 0.0f;

    v2f a; a.x = ax; a.y = ay;
    v2f b; b.x = bx; b.y = by;
    acc = __builtin_amdgcn_wmma_f32_16x16x4_f32(false, a, false, b,
                                                (short)0, acc, false, false);
  }
  return acc;
}

// ---------------------------------------------------------------------------
// Fused: out = relu(agg @ Wo.T + bo)  -->  h = out @ Wx.T + bx
// WMMA for the first 9x9 GEMM, LDS bounce (D layout -> row major), scalar
// second 9x9 GEMM per thread. Saves a full [N,9] global round trip.
// ---------------------------------------------------------------------------
__global__ void out_node_fused(const float* __restrict__ agg, // [N,9]
                               const float* __restrict__ Wo,  // [9,9]
                               const float* __restrict__ bo,  // [9]
                               const float* __restrict__ Wx,  // [9,9]
                               const float* __restrict__ bx,  // [9]
                               float* __restrict__ h,         // [N,9]
                               int N) {
  __shared__ float tile[8][16][9]; // 8 waves * 16 rows * 9 cols
  const int lane = threadIdx.x & 31;
  const int wave = threadIdx.x >> 5;
  const long long base = ((long long)blockIdx.x * 8 + wave) * 16;

  if (base < N) {  // wave-uniform: EXEC all-1s inside for WMMA
    v8f acc = wmma_9in_tile<9>(agg, Wo, base, lane);
    const int n = lane & 15;
    if (n < 9) {
      float bias = bo[n];
#pragma unroll
      for (int v = 0; v < 8; ++v) {
        int r = v + 8 * (lane >> 4);
        tile[wave][r][n] = fmaxf(acc[v] + bias, 0.0f);
      }
    }
  }
  __syncthreads();

  const int tb = threadIdx.x; // node-within-block
  if (tb < 128) {
    long long node = (long long)blockIdx.x * 128 + tb;
    if (node < N) {
      float r[9];
#pragma unroll
      for (int k = 0; k < 9; ++k) r[k] = tile[tb >> 4][tb & 15][k];
#pragma unroll
      for (int j = 0; j < 9; ++j) {
        float a = bx[j];
#pragma unroll
        for (int k = 0; k < 9; ++k) a = fmaf(r[k], Wx[j * 9 + k], a);
        h[node * 9 + j] = a;
      }
    }
  }
}

// ---------------------------------------------------------------------------
// Final conv output: out = relu(agg @ Wo3.T + bo3), OUTD=4
// ---------------------------------------------------------------------------
__global__ void out_linear_wmma4(const float* __restrict__ agg, // [N,9]
                                 const float* __restrict__ Wo,  // [4,9]
                                 const float* __restrict__ bo,  // [4]
                                 float* __restrict__ out,       // [N,4]
                                 int N) {
  const int lane = threadIdx.x & 31;
  const int wave = threadIdx.x >> 5;
  const long long base = ((long long)blockIdx.x * (blockDim.x >> 5) + wave) * 16;
  if (base >= N) return;

  v8f acc = wmma_9in_tile<4>(agg, Wo, base, lane);
  const int n = lane & 15;
  float bias = (n < 4) ? bo[n] : 0.0f;
#pragma unroll
  for (int v = 0; v < 8; ++v) {
    long long r = base + v + 8 * (lane >> 4);
    if (n < 4 && r < (long long)N) {
      out[r * 4 + n] = fmaxf(acc[v] + bias, 0.0f);
    }
  }
}

// ---------------------------------------------------------------------------
// Head: z1 = relu([T,24] @ W1.T + b1) via WMMA f32 16x16x4 (6 K-chunks),
// bounce z1 through LDS, then per-thread layer2 (8->4) + log_softmax.
// ---------------------------------------------------------------------------
__global__ void head_kernel(const float* __restrict__ feat, // [T,24]
                            const float* __restrict__ W1,   // [8,24]
                            const float* __restrict__ b1,   // [8]
                            const float* __restrict__ W2,   // [4,8]
                            const float* __restrict__ b2,   // [4]
                            float* __restrict__ out,        // [T,4]
                            int T) {
  __shared__ float z1s[8 * 16 * 8]; // 8 waves * 16 rows * 8 cols
  const int lane = threadIdx.x & 31;
  const int wave = threadIdx.x >> 5;
  const long long base = ((long long)blockIdx.x * 8 + wave) * 16;

  if (base < T) {  // wave-uniform
    const int n  = lane & 15;
    const int kb = (lane >> 4) * 2;
    const int nn = (n < 8) ? n : 7;  // clamp for branchless weight load
    const long long row = base + (lane & 15); // T%16==0 -> in range
    v8f acc = {};
#pragma unroll
    for (int c = 0; c < 6; ++c) {
      const int k0 = c * 4 + kb;
      v2f a; a.x = feat[row * 24 + k0]; a.y = feat[row * 24 + k0 + 1];
      float bxv = W1[nn * 24 + k0];
      float byv = W1[nn * 24 + k0 + 1];
      v2f b; b.x = (n < 8) ? bxv : 0.0f; b.y = (n < 8) ? byv : 0.0f;
      acc = __builtin_amdgcn_wmma_f32_16x16x4_f32(false, a, false, b,
                                                  (short)0, acc, false, false);
    }
    if (n < 8) {
      float bias = b1[n];
#pragma unroll
      for (int v = 0; v < 8; ++v) {
        int r = v + 8 * (lane >> 4);
        z1s[wave * 128 + r * 8 + n] = fmaxf(acc[v] + bias, 0.0f);
      }
    }
  }
  __syncthreads();

  const int tb = threadIdx.x; // tracklet-within-block = wave*16 + row
  if (tb < 128) {
    long long t = (long long)blockIdx.x * 128 + tb;
    if (t < T) {
      const float* z = &z1s[tb * 8];
      float z2[4];
#pragma unroll
      for (int j = 0; j < 4; ++j) {
        float a = b2[j];
#pragma unroll
        for (int k = 0; k < 8; ++k) a = fmaf(z[k], W2[j * 8 + k], a);
        z2[j] = a;
      }
      float mx = fmaxf(fmaxf(z2[0], z2[1]), fmaxf(z2[2], z2[3]));
      float s = 0.0f;
#pragma unroll
      for (int j = 0; j < 4; ++j) s += __expf(z2[j] - mx);
      float lse = mx + __logf(s);
      float4 o;
      o.x = z2[0] - lse; o.y = z2[1] - lse; o.z = z2[2] - lse; o.w = z2[3] - lse;
      *(float4*)&out[t * 4] = o;
    }
  }
}

// ---------------------------------------------------------------------------
// Orchestration: ping-pong two [N,9] f32 buffers in d_ws (108 MB total).
//   node_linear(x->A)
//   per layer: zero(B); edge_scatter(A->B); fused out+next-node (B->A)
//   final: out_linear4(B->A); head(A->d_out)
// ---------------------------------------------------------------------------
extern "C" void kernel_launch(void* const* d_in, const int* in_sizes, int n_in,
                              void* d_out, int out_size, void* d_ws, size_t ws_size,
                              hipStream_t stream) {
  const float* x   = (const float*)d_in[0];
  const int*   ei  = (const int*)d_in[1];
  const float* ea  = (const float*)d_in[2];
  const float* Wx1 = (const float*)d_in[3],  *bx1 = (const float*)d_in[4];
  const float* We1 = (const float*)d_in[5],  *be1 = (const float*)d_in[6];
  const float* Wo1 = (const float*)d_in[7],  *bo1 = (const float*)d_in[8];
  const float* Wx2 = (const float*)d_in[9],  *bx2 = (const float*)d_in[10];
  const float* We2 = (const float*)d_in[11], *be2 = (const float*)d_in[12];
  const float* Wo2 = (const float*)d_in[13], *bo2 = (const float*)d_in[14];
  const float* Wx3 = (const float*)d_in[15], *bx3 = (const float*)d_in[16];
  const float* We3 = (const float*)d_in[17], *be3 = (const float*)d_in[18];
  const float* Wo3 = (const float*)d_in[19], *bo3 = (const float*)d_in[20];
  const float* W1  = (const float*)d_in[21], *b1  = (const float*)d_in[22];
  const float* W2  = (const float*)d_in[23], *b2  = (const float*)d_in[24];
  float* out = (float*)d_out;

  const int N = in_sizes[0];        // 1,500,000
  const int E = in_sizes[2] / 2;    // 12,000,000
  const int T = N / 6;              // 250,000
  const long long N9 = (long long)N * 9;

  float* A = (float*)d_ws;          // [N,9]
  float* B = A + N9;                // [N,9]

  const int TPB = 256;
  dim3 blk(TPB);
  unsigned gN = (unsigned)((N + TPB - 1) / TPB);
  unsigned gE = (unsigned)((E + TPB - 1) / TPB);
  unsigned gW = (unsigned)(((long long)N + 127) / 128); // 16 nodes/wave, 8 waves/blk
  unsigned gH = (unsigned)((T + 127) / 128);
  unsigned gZ = 2048;

  // conv1 front: h1 = x@Wx1.T+bx1 -> A
  node_linear<1><<<gN, blk, 0, stream>>>(x, Wx1, bx1, A, N);

  // conv1 scatter + fused (out1 -> h2) -> A
  zero_f32<<<gZ, blk, 0, stream>>>(B, N9);
  edge_scatter<<<gE, blk, 0, stream>>>(ei, ea, We1, be1, A, B, E);
  out_node_fused<<<gW, blk, 0, stream>>>(B, Wo1, bo1, Wx2, bx2, A, N);

  // conv2 scatter + fused (out2 -> h3) -> A
  zero_f32<<<gZ, blk, 0, stream>>>(B, N9);
  edge_scatter<<<gE, blk, 0, stream>>>(ei, ea, We2, be2, A, B, E);
  out_node_fused<<<gW, blk, 0, stream>>>(B, Wo2, bo2, Wx3, bx3, A, N);

  // conv3 scatter + final out [N,4] -> A
  zero_f32<<<gZ, blk, 0, stream>>>(B, N9);
  edge_scatter<<<gE, blk, 0, stream>>>(ei, ea, We3, be3, A, B, E);
  out_linear_wmma4<<<gW, blk, 0, stream>>>(B, Wo3, bo3, A, N);

  // head: A viewed as [T,24]
  head_kernel<<<gH, blk, 0, stream>>>(A, W1, b1, W2, b2, out, T);
}